// AdExNeuron_52415780880475
// MI455X (gfx1250) — compile-verified
//
#include <hip/hip_runtime.h>
#include <stdint.h>

// AdEx neuron scan on MI455X (gfx1250).
// 32768 independent chains of length 1024; pure HBM-streaming problem
// (~268 MB total -> ~11.5 us floor at 23.3 TB/s). Input staged through LDS
// with the CDNA5 Tensor Data Mover (tensor_load_to_lds), 4-deep pipelined
// (256 blocks x 3 x 16 KB ~= 12 MB in flight ~ bandwidth-delay product),
// synchronized with s_wait_tensorcnt + workgroup barriers.

#define B_DIM   32
#define C_DIM   1024
#define T_DIM   1024
#define BLOCK_T 128                 // t-strip width per block (4 wave32s)
#define KC      32                  // c-steps per TDM tile (16 KB tile)
#define NCHUNK  (C_DIM / KC)        // 32 chunks
#define DEPTH   4                   // TDM pipeline depth (LDS buffers)

typedef __attribute__((ext_vector_type(4))) unsigned int u32x4;
typedef __attribute__((ext_vector_type(8))) unsigned int u32x8;

// Issue one TDM tile load. Group 1 (dims/strides, loop-invariant) prebuilt by
// caller; group 0 (LDS addr + global addr) built per call.
__device__ __forceinline__ void tdm_issue(unsigned lds_addr,
                                          unsigned long long gaddr,
                                          const u32x8& g1,
                                          const u32x4& gz)
{
    u32x4 g0;
    g0[0] = 1u;                                            // count=1, user D#
    g0[1] = lds_addr;                                      // LDS byte address
    g0[2] = (unsigned)(gaddr & 0xFFFFFFFFull);             // global_addr[31:0]
    g0[3] = (unsigned)((gaddr >> 32) & 0x01FFFFFFull)      // global_addr[56:32]
          | (2u << 30);                                    // type = 2 (image)

    asm volatile("tensor_load_to_lds %0, %1, %2, %3"
                 :: "s"(g0), "s"(g1), "s"(gz), "s"(gz)
                 : "memory");
}

__global__ __launch_bounds__(BLOCK_T)
void adex_tdm_kernel(const float* __restrict__ I,
                     const float* __restrict__ p,
                     float* __restrict__ out)
{
    __shared__ float buf[DEPTH][KC * BLOCK_T];   // 4 x 16 KB pipeline buffers

    const int tid   = threadIdx.x;
    const int wave  = tid >> 5;
    const int strip = blockIdx.x & (T_DIM / BLOCK_T - 1);  // 0..7
    const int b     = blockIdx.x >> 3;
    const int t     = strip * BLOCK_T + tid;

    // AdEx parameters
    const float tau_m  = p[0], E_L = p[1], V_T = p[2], Delta_T = p[3], R = p[4];
    const float tau_w  = p[5], a_p = p[6], b_spk = p[7];
    const float V_reset = p[8], V_spike = p[9], dt = p[10];
    const float k_m    = dt / tau_m;
    const float k_w    = dt / tau_w;
    const float inv_DT = 1.0f / Delta_T;

    float V = E_L;
    float w = 0.0f;

    // Uniform tile base for this block (c = 0); stride between chunks = KC rows
    const unsigned long long gbase =
        (unsigned long long)(uintptr_t)I +
        (((unsigned long long)b * C_DIM) * T_DIM +
          (unsigned long long)strip * BLOCK_T) * 4ull;
    const unsigned long long gchunk = (unsigned long long)KC * T_DIM * 4ull;

    // Low 32 bits of a flat shared pointer == LDS byte offset (aperture in hi)
    unsigned ldsb[DEPTH];
    #pragma unroll
    for (int i = 0; i < DEPTH; ++i)
        ldsb[i] = (unsigned)(uintptr_t)(&buf[i][0]);

    // ---- loop-invariant descriptor words (D# group 1; groups 2/3 zero) ----
    const unsigned long long s1e = (unsigned long long)C_DIM * T_DIM; // dim1 stride
    u32x8 g1;
    g1[0] = (2u << 16);                                    // data_size=4B, no mcast
    g1[1] = ((unsigned)T_DIM & 0xFFFFu) << 16;             // tensor_dim0[15:0]
    g1[2] = (((unsigned)T_DIM >> 16) & 0xFFFFu)            // tensor_dim0[31:16]
          | (((unsigned)C_DIM & 0xFFFFu) << 16);           // tensor_dim1[15:0]
    g1[3] = (((unsigned)C_DIM >> 16) & 0xFFFFu)            // tensor_dim1[31:16]
          | ((unsigned)BLOCK_T << 16);                     // tile_dim0 = 128
    g1[4] = (unsigned)KC;                                  // tile_dim1 = 32
    g1[5] = (unsigned)T_DIM;                               // dim0_stride[31:0]
    g1[6] = ((unsigned)(s1e & 0xFFFFull) << 16);           // dim1_stride[15:0]
    g1[7] = (unsigned)((s1e >> 16) & 0xFFFFFFFFull);       // dim1_stride[47:16]
    const u32x4 gz = {0u, 0u, 0u, 0u};

    // ---- prime the pipeline: chunks 0 .. DEPTH-2 ----
    if (wave == 0) {
        #pragma unroll
        for (int k = 0; k < DEPTH - 1; ++k)
            tdm_issue(ldsb[k], gbase + (unsigned long long)k * gchunk, g1, gz);
    }

    float* o = out + ((unsigned long long)b * C_DIM) * T_DIM + t;

    for (int n = 0; n < NCHUNK; ++n) {
        if (wave == 0) {
            const int k = n + DEPTH - 1;               // chunk to issue
            if (k < NCHUNK) {
                tdm_issue(ldsb[k & (DEPTH - 1)], gbase + (unsigned long long)k * gchunk,
                          g1, gz);
                // TDM ops from one wave complete in order:
                // <= DEPTH-1 outstanding  =>  chunk n has landed in LDS
                __builtin_amdgcn_s_wait_tensorcnt(DEPTH - 1);
            } else {
                __builtin_amdgcn_s_wait_tensorcnt(0);  // drain tail
            }
        }
        __syncthreads();                               // chunk n visible to block

        const float* lbuf = &buf[n & (DEPTH - 1)][0];
        const int cbase = n * KC;
        #pragma unroll 4
        for (int cc = 0; cc < KC; ++cc) {
            const float Iv = lbuf[cc * BLOCK_T + tid];
            const float ex = Delta_T * __expf((V - V_T) * inv_DT);
            V += k_m * ((E_L - V) + ex + R * (Iv - w));
            w += k_w * (a_p * (V - E_L) - w);
            float spk;
            if (V >= V_spike) { spk = 1.0f; V = V_reset; w += b_spk; }
            else              { spk = 0.0f; }
            o[(unsigned long long)(cbase + cc) * T_DIM] = spk;   // coalesced
        }
        __syncthreads();           // all waves done reading before buffer reuse
    }
}

extern "C" void kernel_launch(void* const* d_in, const int* in_sizes, int n_in,
                              void* d_out, int out_size, void* d_ws, size_t ws_size,
                              hipStream_t stream) {
    const float* I = (const float*)d_in[0];     // [32, 1024, 1024] f32
    const float* p = (const float*)d_in[1];     // 11 params f32
    float* out = (float*)d_out;                 // [32, 1024, 1024] f32

    dim3 grid(B_DIM * (T_DIM / BLOCK_T));       // 256 blocks
    dim3 block(BLOCK_T);                        // 128 threads = 4 wave32s
    hipLaunchKernelGGL(adex_tdm_kernel, grid, block, 0, stream, I, p, out);
}